// QAGNN_86337432584685
// MI455X (gfx1250) — compile-verified
//
#include <hip/hip_runtime.h>
#include <math.h>
#include <stdint.h>

typedef __attribute__((ext_vector_type(2))) float v2f;
typedef __attribute__((ext_vector_type(8))) float v8f;
typedef __attribute__((ext_vector_type(4))) int   v4i;

#define N_GRAPHS 64
#define NPG 200                 // nodes per graph
#define NN (N_GRAPHS * NPG)     // 12800
#define EE 128000
#define LMD 1024
#define HD 128
#define NET 42                  // num edge types
#define HOPS 5
#define NEG_SLOPE 0.2f
#define LCH 4                   // L-chunks for k_bil_A

#if defined(__HIP_DEVICE_COMPILE__) && defined(__has_builtin)
#if __has_builtin(__builtin_amdgcn_global_load_async_to_lds_b128) && \
    __has_builtin(__builtin_amdgcn_s_wait_asynccnt)
#define USE_ASYNC_LDS 1
#endif
#endif

typedef __attribute__((address_space(1))) v4i* gptr_v4i;
typedef __attribute__((address_space(3))) v4i* lptr_v4i;

static __device__ __forceinline__ int imin(int a, int b) { return a < b ? a : b; }

// hardware FP32 atomic add (global_atomic_add_f32) where available
static __device__ __forceinline__ void atomAddF(float* p, float v) {
#if defined(__HIP_PLATFORM_AMD__)
    unsafeAtomicAdd(p, v);
#else
    atomicAdd(p, v);
#endif
}

// ---------------- monotone float<->uint key for atomicMax on floats ----------
static __device__ __forceinline__ unsigned fkey(float f) {
    unsigned u = __float_as_uint(f);
    return (u & 0x80000000u) ? ~u : (u | 0x80000000u);
}
static __device__ __forceinline__ float unfkey(unsigned k) {
    return __uint_as_float((k & 0x80000000u) ? (k & 0x7fffffffu) : ~k);
}

// ---------------- generic strided 16x16 FP32 WMMA tile ----------------------
// C(16x16) = A(16xK) * B(Kx16); A[m,k] = A[m*rsA + k*csA]; B[k,n] = B[k*rsB + n*csB]
static __device__ __forceinline__ v8f wmma_tile_f32(
        const float* __restrict__ A, int rsA, int csA, int mValid,
        const float* __restrict__ B, int rsB, int csB, int nValid, int K) {
    int lane = threadIdx.x & 31;
    int hh = lane >> 4;          // K-pair selector (ISA A/B layout for 32-bit)
    int id = lane & 15;
    int ma = id < mValid ? id : (mValid - 1);   // clamp OOB loads; stores predicated
    int na = id < nValid ? id : (nValid - 1);
    v8f acc = {};
    for (int k0 = 0; k0 < K; k0 += 4) {
        int ka = k0 + hh * 2;
        v2f a, b;
        a.x = A[(size_t)ma * rsA + (size_t)ka * csA];
        a.y = A[(size_t)ma * rsA + (size_t)(ka + 1) * csA];
        b.x = B[(size_t)ka * rsB + (size_t)na * csB];
        b.y = B[(size_t)(ka + 1) * rsB + (size_t)na * csB];
        acc = __builtin_amdgcn_wmma_f32_16x16x4_f32(false, a, false, b,
                                                    (short)0, acc, false, false);
    }
    return acc;
}

static __device__ __forceinline__ void wmma_store(
        float* __restrict__ C, int ldc, v8f acc, int mValid, int nValid,
        const float* __restrict__ bias /* pre-offset by n0, or nullptr */) {
    int lane = threadIdx.x & 31;
    int hh = lane >> 4, id = lane & 15;
    if (id >= nValid) return;
    float bv = bias ? bias[id] : 0.0f;
#pragma unroll
    for (int r = 0; r < 8; ++r) {
        int m = r + hh * 8;
        if (m < mValid) C[(size_t)m * ldc + id] = acc[r] + bv;
    }
}

// ---------------- stage 1: ctx_emb = lm_context @ W_lm + b_lm  [64,128] -----
__global__ void k_ctx_gemm(const float* __restrict__ lm, const float* __restrict__ Wlm,
                           const float* __restrict__ blm, float* __restrict__ ctx) {
    int mt = blockIdx.x;  // 0..3
    int nt = blockIdx.y;  // 0..7
    v8f acc = wmma_tile_f32(lm + (size_t)mt * 16 * LMD, LMD, 1, 16,
                            Wlm + nt * 16, HD, 1, 16, LMD);
    wmma_store(ctx + (size_t)mt * 16 * HD + nt * 16, HD, acc, 16, 16, blm + nt * 16);
}

// ---------------- stage 2: x2 = node_emb with ctx rows injected -------------
__global__ void k_build_x2(const float* __restrict__ node_emb,
                           const float* __restrict__ ctx, float* __restrict__ x2) {
    int i = blockIdx.x * blockDim.x + threadIdx.x;
    if (i >= NN * HD) return;
    int n = i / HD, h = i - n * HD;
    float v = node_emb[i];
    if (n % NPG == 0) v = ctx[(n / NPG) * HD + h];
    x2[i] = v;
}

// ---------------- stage 3: A[g,k,h] += sum_{l in chunk} lm[g,l]*W_bil[k,l,h] -
// One wave: one (k, h-tile, L-chunk), all four g-tiles -> W_bil streamed once,
// 4096 waves to keep 23.3 TB/s HBM busy. Abuf must be pre-zeroed.
__global__ void k_bil_A(const float* __restrict__ lm, const float* __restrict__ Wbil,
                        float* __restrict__ Abuf) {
    int ht = blockIdx.x;  // 0..7
    int k  = blockIdx.y;  // 0..127
    int lc = blockIdx.z;  // 0..LCH-1
    int lane = threadIdx.x & 31;
    int hh = lane >> 4, id = lane & 15;
    const int L0 = lc * (LMD / LCH), L1 = L0 + (LMD / LCH);
    const float* Bbase = Wbil + (size_t)k * LMD * HD + ht * 16;  // B[l,h]=base[l*HD+h]
    v8f acc[4] = {v8f{}, v8f{}, v8f{}, v8f{}};
    for (int l0 = L0; l0 < L1; l0 += 4) {
        int ka = l0 + hh * 2;
        __builtin_prefetch(Bbase + (size_t)(ka + 32) * HD + id, 0, 1);
        v2f b;
        b.x = Bbase[(size_t)ka * HD + id];
        b.y = Bbase[(size_t)(ka + 1) * HD + id];
#pragma unroll
        for (int t = 0; t < 4; ++t) {
            v2f a = *(const v2f*)(lm + (size_t)(t * 16 + id) * LMD + ka);
            acc[t] = __builtin_amdgcn_wmma_f32_16x16x4_f32(false, a, false, b,
                                                           (short)0, acc[t], false, false);
        }
    }
#pragma unroll
    for (int t = 0; t < 4; ++t)
#pragma unroll
        for (int r = 0; r < 8; ++r) {
            int g = t * 16 + r + hh * 8;
            atomAddF(&Abuf[(size_t)g * HD * HD + (size_t)k * HD + ht * 16 + id], acc[t][r]);
        }
}

// ---------------- stage 4: nodes = X2_g @ A_g^T + b_bil ---------------------
__global__ void k_bil_apply(const float* __restrict__ x2, const float* __restrict__ Abuf,
                            const float* __restrict__ bbil, float* __restrict__ nodes) {
    int mt = blockIdx.x;  // 0..12
    int kt = blockIdx.y;  // 0..7
    int g  = blockIdx.z;  // 0..63
    int m0 = mt * 16;
    int mValid = imin(16, NPG - m0);
    const float* A = x2 + (size_t)(g * NPG + m0) * HD;
    const float* B = Abuf + (size_t)g * HD * HD + kt * 16;  // B[h,kc]=base[kc*HD+h]
    v8f acc = wmma_tile_f32(A, HD, 1, mValid, B, 1, HD, 16, HD);
    wmma_store(nodes + (size_t)(g * NPG + m0) * HD + kt * 16, HD, acc, mValid, 16,
               bbil + kt * 16);
}

// ---------------- GAT hop: h = nodes @ lin, B-tile staged in LDS ------------
// 8 waves / block: async-copy B[k=0..127][n0..n0+15] (8 KB) to LDS once,
// each wave then computes one 16x16 m-tile reading B from DS.
__global__ void __launch_bounds__(256)
k_hop_gemm(const float* __restrict__ nodes, const float* __restrict__ lin,
           float* __restrict__ hbuf) {
    __shared__ float ldsB[HD * 16];
    int nt = blockIdx.y;
    int n0 = nt * 16;
#if defined(USE_ASYNC_LDS)
    for (int q = threadIdx.x; q < HD * 4; q += 256) {
        const float* gsrc = lin + (size_t)(q >> 2) * HD + n0 + (q & 3) * 4;
        __builtin_amdgcn_global_load_async_to_lds_b128(
            (gptr_v4i)(uintptr_t)gsrc,
            (lptr_v4i)&ldsB[q * 4], 0, 0);
    }
    __builtin_amdgcn_s_wait_asynccnt(0);
    __syncthreads();
#else
    for (int q = threadIdx.x; q < HD * 4; q += 256) {
        *(float4*)&ldsB[q * 4] = *(const float4*)(lin + (size_t)(q >> 2) * HD + n0 + (q & 3) * 4);
    }
    __syncthreads();
#endif
    int mt = blockIdx.x * 8 + (threadIdx.x >> 5);   // 8 m-tiles per block
    int lane = threadIdx.x & 31;
    int hh = lane >> 4, id = lane & 15;
    const float* A = nodes + (size_t)mt * 16 * HD;
    v8f acc = {};
    for (int k0 = 0; k0 < HD; k0 += 4) {
        int ka = k0 + hh * 2;
        v2f a = *(const v2f*)(A + (size_t)id * HD + ka);
        v2f b;
        b.x = ldsB[ka * 16 + id];
        b.y = ldsB[(ka + 1) * 16 + id];
        acc = __builtin_amdgcn_wmma_f32_16x16x4_f32(false, a, false, b,
                                                    (short)0, acc, false, false);
    }
    wmma_store(hbuf + (size_t)mt * 16 * HD + n0, HD, acc, 16, 16, nullptr);
}

// ---------------- per-(hop,edge-type) scalar table --------------------------
// aet[i,t] = edge_emb[t] . (lin_edge[i] @ att_edge[i])
__global__ void k_aet(const float* __restrict__ lin_edge, const float* __restrict__ att_edge,
                      const float* __restrict__ edge_emb, float* __restrict__ aet) {
    int i = blockIdx.x;     // hop
    int t = threadIdx.x;    // 0..127
    __shared__ float v[HD];
    const float* Le = lin_edge + (size_t)i * HD * HD;
    const float* ae = att_edge + (size_t)i * HD;
    float s = 0.0f;
    for (int h = 0; h < HD; ++h) s += Le[(size_t)t * HD + h] * ae[h];
    v[t] = s;
    __syncthreads();
    if (t < NET) {
        float r = 0.0f;
        for (int h2 = 0; h2 < HD; ++h2) r += edge_emb[(size_t)t * HD + h2] * v[h2];
        aet[i * NET + t] = r;
    }
}

// ---------------- misc elementwise ------------------------------------------
__global__ void k_zero_f(float* p, int n) {
    int i = blockIdx.x * blockDim.x + threadIdx.x;
    if (i < n) p[i] = 0.0f;
}
__global__ void k_set_u(unsigned* p, unsigned v, int n) {
    int i = blockIdx.x * blockDim.x + threadIdx.x;
    if (i < n) p[i] = v;
}
__global__ void k_deg(const int* __restrict__ dst, float* __restrict__ deg) {
    int e = blockIdx.x * blockDim.x + threadIdx.x;
    if (e < EE) atomAddF(&deg[dst[e]], 1.0f);
}

// ---------------- per-node attention scalars (one wave per node) ------------
__global__ void k_node_scalars(const float* __restrict__ hbuf,
                               const float* __restrict__ wsrc, const float* __restrict__ wdst,
                               float* __restrict__ a_src, float* __restrict__ a_dst) {
    int n = blockIdx.x;
    int lane = threadIdx.x;
    const float* hr = hbuf + (size_t)n * HD;
    float s1 = 0.0f, s2 = 0.0f;
    for (int j = lane; j < HD; j += 32) {
        float hv = hr[j];
        s1 += hv * wsrc[j];
        s2 += hv * wdst[j];
    }
#pragma unroll
    for (int off = 16; off > 0; off >>= 1) {
        s1 += __shfl_xor(s1, off, 32);
        s2 += __shfl_xor(s2, off, 32);
    }
    if (lane == 0) { a_src[n] = s1; a_dst[n] = s2; }
}

// ---------------- edge logits + a_loop accumulation + segment max -----------
__global__ void k_edge_logits(const int* __restrict__ src, const int* __restrict__ dst,
                              const int* __restrict__ etype, const float* __restrict__ aet_i,
                              const float* __restrict__ a_src, const float* __restrict__ a_dst,
                              float* __restrict__ logits, float* __restrict__ aloop_sum,
                              unsigned* __restrict__ mkey) {
    int e = blockIdx.x * blockDim.x + threadIdx.x;
    if (e >= EE) return;
    int s = src[e], d = dst[e];
    float ae = aet_i[etype[e]];
    atomAddF(&aloop_sum[d], ae);
    float x = a_src[s] + a_dst[d] + ae;
    float lg = x > 0.0f ? x : NEG_SLOPE * x;
    logits[e] = lg;
    atomicMax(&mkey[d], fkey(lg));
}

__global__ void k_self_logits(const float* __restrict__ a_src, const float* __restrict__ a_dst,
                              const float* __restrict__ aloop_sum, const float* __restrict__ deg,
                              float* __restrict__ logits, unsigned* __restrict__ mkey) {
    int n = blockIdx.x * blockDim.x + threadIdx.x;
    if (n >= NN) return;
    float al = aloop_sum[n] / fmaxf(deg[n], 1.0f);
    float x = a_src[n] + a_dst[n] + al;
    float lg = x > 0.0f ? x : NEG_SLOPE * x;
    logits[EE + n] = lg;
    atomicMax(&mkey[n], fkey(lg));
}

// ---------------- exp + segment denominator ---------------------------------
__global__ void k_expsum(const int* __restrict__ dst, const float* __restrict__ logits,
                         const unsigned* __restrict__ mkey, float* __restrict__ pbuf,
                         float* __restrict__ denom) {
    int e = blockIdx.x * blockDim.x + threadIdx.x;
    if (e >= EE + NN) return;
    int d = (e < EE) ? dst[e] : (e - EE);
    float p = expf(logits[e] - unfkey(mkey[d]));
    pbuf[e] = p;
    atomAddF(&denom[d], p);
}

// ---------------- weighted scatter aggregation ------------------------------
__global__ void k_aggregate(const int* __restrict__ src, const int* __restrict__ dst,
                            const float* __restrict__ pbuf, const float* __restrict__ denom,
                            const float* __restrict__ hbuf, float* __restrict__ outacc) {
    int e = blockIdx.x;     // 0 .. EE+NN-1
    int t = threadIdx.x;    // 0..127
    int s, d;
    if (e < EE) { s = src[e]; d = dst[e]; }
    else        { s = d = e - EE; }
    float w = pbuf[e] / denom[d];
    atomAddF(&outacc[(size_t)d * HD + t], w * hbuf[(size_t)s * HD + t]);
}

// ---------------- bias + exact-erf GELU -> nodes ----------------------------
__global__ void k_gelu(const float* __restrict__ outacc, const float* __restrict__ bias_i,
                       float* __restrict__ nodes) {
    int i = blockIdx.x * blockDim.x + threadIdx.x;
    if (i >= NN * HD) return;
    float x = outacc[i] + bias_i[i % HD];
    nodes[i] = 0.5f * x * (1.0f + erff(x * 0.70710678118654752f));
}

// ---------------- gather context rows ---------------------------------------
__global__ void k_gather(const float* __restrict__ nodes, float* __restrict__ out) {
    int i = blockIdx.x * blockDim.x + threadIdx.x;
    if (i >= N_GRAPHS * HD) return;
    int g = i / HD, h = i - g * HD;
    out[i] = nodes[(size_t)(g * NPG) * HD + h];
}

// ============================================================================
extern "C" void kernel_launch(void* const* d_in, const int* in_sizes, int n_in,
                              void* d_out, int out_size, void* d_ws, size_t ws_size,
                              hipStream_t stream) {
    const float* lm_context = (const float*)d_in[0];
    const float* node_emb   = (const float*)d_in[1];
    const float* W_lm       = (const float*)d_in[2];
    const float* b_lm       = (const float*)d_in[3];
    const float* W_bil      = (const float*)d_in[4];
    const float* b_bil      = (const float*)d_in[5];
    const float* edge_emb   = (const float*)d_in[6];
    const float* gat_lin    = (const float*)d_in[7];
    const float* att_src    = (const float*)d_in[8];
    const float* att_dst    = (const float*)d_in[9];
    const float* lin_edge   = (const float*)d_in[10];
    const float* att_edge   = (const float*)d_in[11];
    const float* gat_bias   = (const float*)d_in[12];
    const int*   edge_index = (const int*)d_in[13];
    const int*   edge_type  = (const int*)d_in[14];
    // d_in[15]=batch, d_in[16]=ctx_nodes: layout is structured, not needed.
    const int* src = edge_index;
    const int* dst = edge_index + EE;
    float* out = (float*)d_out;

    // workspace carve-up (floats)
    float* w = (float*)d_ws;
    float* ctx      = w;               w += N_GRAPHS * HD;
    float* x2       = w;               w += (size_t)NN * HD;
    float* Abuf     = w;               w += (size_t)N_GRAPHS * HD * HD;
    float* nodes    = w;               w += (size_t)NN * HD;
    float* hbuf     = w;               w += (size_t)NN * HD;
    float* outacc   = w;               w += (size_t)NN * HD;
    float* a_src    = w;               w += NN;
    float* a_dst    = w;               w += NN;
    float* aloop    = w;               w += NN;
    float* deg      = w;               w += NN;
    float* denom    = w;               w += NN;
    unsigned* mkey  = (unsigned*)w;    w += NN;
    float* logits   = w;               w += (EE + NN);
    float* pbuf     = w;               w += (EE + NN);
    float* aet      = w;               w += HOPS * NET;

    const int BLK = 256;
    auto cdiv = [](int a, int b) { return (a + b - 1) / b; };

    // ---- pre-GAT pipeline ----
    k_ctx_gemm<<<dim3(4, 8), 32, 0, stream>>>(lm_context, W_lm, b_lm, ctx);
    k_build_x2<<<cdiv(NN * HD, BLK), BLK, 0, stream>>>(node_emb, ctx, x2);
    k_zero_f<<<cdiv(N_GRAPHS * HD * HD, BLK), BLK, 0, stream>>>(Abuf, N_GRAPHS * HD * HD);
    k_bil_A<<<dim3(8, 128, LCH), 32, 0, stream>>>(lm_context, W_bil, Abuf);
    k_bil_apply<<<dim3(13, 8, N_GRAPHS), 32, 0, stream>>>(x2, Abuf, b_bil, nodes);

    // ---- hop-invariant precomputation ----
    k_zero_f<<<cdiv(NN, BLK), BLK, 0, stream>>>(deg, NN);
    k_deg<<<cdiv(EE, BLK), BLK, 0, stream>>>(dst, deg);
    k_aet<<<HOPS, HD, 0, stream>>>(lin_edge, att_edge, edge_emb, aet);

    // ---- GAT hops ----
    for (int i = 0; i < HOPS; ++i) {
        k_hop_gemm<<<dim3(NN / 16 / 8, 8), 256, 0, stream>>>(
            nodes, gat_lin + (size_t)i * HD * HD, hbuf);
        k_node_scalars<<<NN, 32, 0, stream>>>(hbuf, att_src + (size_t)i * HD,
                                              att_dst + (size_t)i * HD, a_src, a_dst);
        k_zero_f<<<cdiv(NN, BLK), BLK, 0, stream>>>(aloop, NN);
        k_zero_f<<<cdiv(NN, BLK), BLK, 0, stream>>>(denom, NN);
        k_zero_f<<<cdiv(NN * HD, BLK), BLK, 0, stream>>>(outacc, NN * HD);
        k_set_u<<<cdiv(NN, BLK), BLK, 0, stream>>>(mkey, 0u, NN);
        k_edge_logits<<<cdiv(EE, BLK), BLK, 0, stream>>>(src, dst, edge_type, aet + i * NET,
                                                         a_src, a_dst, logits, aloop, mkey);
        k_self_logits<<<cdiv(NN, BLK), BLK, 0, stream>>>(a_src, a_dst, aloop, deg, logits, mkey);
        k_expsum<<<cdiv(EE + NN, BLK), BLK, 0, stream>>>(dst, logits, mkey, pbuf, denom);
        k_aggregate<<<EE + NN, HD, 0, stream>>>(src, dst, pbuf, denom, hbuf, outacc);
        k_gelu<<<cdiv(NN * HD, BLK), BLK, 0, stream>>>(outacc, gat_bias + (size_t)i * HD, nodes);
    }

    k_gather<<<cdiv(N_GRAPHS * HD, BLK), BLK, 0, stream>>>(nodes, out);
}